// NetPredictor_33260226740762
// MI455X (gfx1250) — compile-verified
//
#include <hip/hip_runtime.h>

// ---------------------------------------------------------------------------
// NetPredictor GNN for MI455X (gfx1250, wave32, WMMA).
// fp32 end-to-end (matches reference numerics); dense GEMMs use
// V_WMMA_F32_16X16X4_F32 with the per-wave W column-slab held in registers
// and reused across 5 row-tiles per block (5x less L2 weight traffic).
// Sparse mean-aggregation is gather-based over a per-call-built CSR
// (no per-channel atomics; wave32 __shfl edge broadcast).
// ---------------------------------------------------------------------------

typedef __attribute__((ext_vector_type(2))) float v2f;
typedef __attribute__((ext_vector_type(8))) float v8f;

#define HID 128
static constexpr int N_PIN  = 200000;
static constexpr int N_NET  = 50000;
static constexpr int E_P2N  = 400000;
static constexpr int E_N2N  = 800000;
static constexpr int PIN_IN = 16;
static constexpr int NL     = 3;
static constexpr int RTILES = 5;          // row-tiles (16 rows each) per block
static constexpr float LN_EPS = 1e-5f;

// =========================== CSR construction ==============================

__global__ void count_kernel(const int* __restrict__ dst, int E,
                             int* __restrict__ cnt) {
    int t = blockIdx.x * blockDim.x + threadIdx.x;
    if (t < E) atomicAdd(&cnt[dst[t]], 1);
}

// per-block exclusive scan (256 elements / block) + block totals
__global__ __launch_bounds__(256)
void scan_block_kernel(const int* __restrict__ cnt, int n,
                       int* __restrict__ offs, int* __restrict__ bsum) {
    __shared__ int tmp[256];
    const int tid = threadIdx.x;
    const int i   = blockIdx.x * 256 + tid;
    const int v   = (i < n) ? cnt[i] : 0;
    tmp[tid] = v;
    __syncthreads();
    for (int off = 1; off < 256; off <<= 1) {
        int t = (tid >= off) ? tmp[tid - off] : 0;
        __syncthreads();
        tmp[tid] += t;
        __syncthreads();
    }
    if (i < n) offs[i] = tmp[tid] - v;           // exclusive
    if (tid == 255) bsum[blockIdx.x] = tmp[255];
}

__global__ void scan_bsum_kernel(int* __restrict__ bsum, int nb) {
    if (blockIdx.x == 0 && threadIdx.x == 0) {
        int acc = 0;
        for (int i = 0; i < nb; ++i) { int t = bsum[i]; bsum[i] = acc; acc += t; }
    }
}

__global__ void scan_add_kernel(int* __restrict__ offs,
                                const int* __restrict__ bsum, int n) {
    int i = blockIdx.x * blockDim.x + threadIdx.x;
    if (i < n) offs[i] += bsum[i >> 8];
}

__global__ void csr_fill_kernel(const int* __restrict__ src,
                                const int* __restrict__ dst,
                                const float* __restrict__ ew, int E,
                                int* __restrict__ cursor,
                                int* __restrict__ csr_src,
                                float* __restrict__ csr_w) {
    int t = blockIdx.x * blockDim.x + threadIdx.x;
    if (t >= E) return;
    int pos = atomicAdd(&cursor[dst[t]], 1);
    csr_src[pos] = src[t];
    if (csr_w != nullptr) csr_w[pos] = ew[t];
}

// ==================== gather-based mean aggregation ========================
// One wave per destination node; lane handles channels [4*lane, 4*lane+4).
// Edge ids/weights loaded 32-at-a-time and broadcast via __shfl (wave32).
__global__ __launch_bounds__(256)
void gather_kernel(const float* __restrict__ hsrc,
                   const int* __restrict__ offs, const int* __restrict__ cnt,
                   const int* __restrict__ csr_src,
                   const float* __restrict__ csr_w,
                   float* __restrict__ agg) {
    const int lane = threadIdx.x & 31;
    const int node = blockIdx.x * 8 + (threadIdx.x >> 5);
    const int base = offs[node];
    const int c    = cnt[node];
    const int ch   = lane * 4;

    float ax = 0.f, ay = 0.f, az = 0.f, aw = 0.f;
    if (csr_w == nullptr) {
        for (int i0 = 0; i0 < c; i0 += 32) {
            int myid = (i0 + lane < c) ? csr_src[base + i0 + lane] : 0;
            int kmax = c - i0; kmax = kmax < 32 ? kmax : 32;
            for (int j = 0; j < kmax; ++j) {
                int s = __shfl(myid, j, 32);
                const float4 v = *(const float4*)(hsrc + (long)s * HID + ch);
                ax += v.x; ay += v.y; az += v.z; aw += v.w;
            }
        }
    } else {
        for (int i0 = 0; i0 < c; i0 += 32) {
            int   myid = (i0 + lane < c) ? csr_src[base + i0 + lane] : 0;
            float myw  = (i0 + lane < c) ? csr_w[base + i0 + lane] : 0.f;
            int kmax = c - i0; kmax = kmax < 32 ? kmax : 32;
            for (int j = 0; j < kmax; ++j) {
                int   s = __shfl(myid, j, 32);
                float w = __shfl(myw, j, 32);
                const float4 v = *(const float4*)(hsrc + (long)s * HID + ch);
                ax += v.x * w; ay += v.y * w; az += v.z * w; aw += v.w * w;
            }
        }
    }
    const float inv = 1.0f / (float)(c < 1 ? 1 : c);   // mean with clamp(deg,1)
    float4 r; r.x = ax * inv; r.y = ay * inv; r.z = az * inv; r.w = aw * inv;
    *(float4*)(agg + (long)node * HID + ch) = r;
}

// ===================== dense stages (WMMA fp32) ============================

// projection: h_pin = relu(X @ Wproj + b) ; X: [N_PIN,16], W: [16,128]
__global__ __launch_bounds__(256)
void proj_kernel(const float* __restrict__ X, const float* __restrict__ W,
                 const float* __restrict__ bias, float* __restrict__ out) {
    __shared__ float At[16 * PIN_IN];
    const int tid  = threadIdx.x;
    const int row0 = blockIdx.x * 16;

    At[tid] = X[(long)row0 * PIN_IN + tid];   // 16x16 tile, contiguous
    __syncthreads();

    const int wave = tid >> 5, lane = tid & 31;
    const int m = lane & 15, half = lane >> 4, koff = half * 2;
    const int col = wave * 16 + m;

    v8f acc = {};
#pragma unroll
    for (int kb = 0; kb < PIN_IN; kb += 4) {
        v2f a, b;
        a.x = At[m * PIN_IN + kb + koff];
        a.y = At[m * PIN_IN + kb + koff + 1];
        b.x = W[(kb + koff) * HID + col];
        b.y = W[(kb + koff + 1) * HID + col];
        acc = __builtin_amdgcn_wmma_f32_16x16x4_f32(false, a, false, b,
                                                    (short)0, acc, false, false);
    }
    const float bb = bias[col];
#pragma unroll
    for (int i = 0; i < 8; ++i) {
        int r = i + half * 8;
        out[(long)(row0 + r) * HID + col] = fmaxf(acc[i] + bb, 0.0f);
    }
}

// fused: h_io = relu(LN(h_io + agg @ W + b))   (agg pre-normalized)
// W column-slab (128x16 per wave, 64 VGPRs) preloaded once, reused for
// RTILES row-tiles per block.
__global__ __launch_bounds__(256)
void gconv_finish_kernel(const float* __restrict__ agg,
                         const float* __restrict__ W,
                         const float* __restrict__ bias,
                         float* __restrict__ h_io,
                         const float* __restrict__ lng,
                         const float* __restrict__ lnb) {
    __shared__ float At[16 * HID];
    __shared__ float Xt[16 * HID];
    __shared__ float red[256];
    __shared__ float mu_s[16], rs_s[16];

    const int tid  = threadIdx.x;
    const int wave = tid >> 5, lane = tid & 31;
    const int m = lane & 15, half = lane >> 4, koff = half * 2;
    const int col = wave * 16 + m;
    const int lr = tid >> 4, sc = (tid & 15) * 8;

    // preload this wave's 128x16 W slab into registers (reused by all tiles)
    v2f bf[32];
#pragma unroll
    for (int kk = 0; kk < 32; ++kk) {
        const int kb = kk * 4;
        bf[kk].x = W[(kb + koff) * HID + col];
        bf[kk].y = W[(kb + koff + 1) * HID + col];
    }
    const float bb = bias[col];

    for (int t = 0; t < RTILES; ++t) {
        const int row0 = (blockIdx.x * RTILES + t) * 16;

#pragma unroll
        for (int j = 0; j < 8; ++j) {
            int e = tid + j * 256;
            At[e] = agg[(long)row0 * HID + e];
        }
        __syncthreads();

        v8f acc = {};
#pragma unroll
        for (int kk = 0; kk < 32; ++kk) {
            const int kb = kk * 4;
            v2f a;
            a.x = At[m * HID + kb + koff];
            a.y = At[m * HID + kb + koff + 1];
            acc = __builtin_amdgcn_wmma_f32_16x16x4_f32(
                false, a, false, bf[kk], (short)0, acc, false, false);
        }

        // bias + residual into LDS for cross-wave LayerNorm
#pragma unroll
        for (int i = 0; i < 8; ++i) {
            int r = i + half * 8;
            Xt[r * HID + col] = acc[i] + bb + h_io[(long)(row0 + r) * HID + col];
        }
        __syncthreads();

        // LayerNorm: 16 threads per row (row = tid>>4, 8-col chunk per thread)
        float ps = 0.f;
#pragma unroll
        for (int j = 0; j < 8; ++j) ps += Xt[lr * HID + sc + j];
        red[tid] = ps;
        __syncthreads();
        if (tid < 16) {
            float s = 0.f;
            for (int k = 0; k < 16; ++k) s += red[tid * 16 + k];
            mu_s[tid] = s * (1.0f / HID);
        }
        __syncthreads();
        const float mu = mu_s[lr];
        float pv = 0.f;
#pragma unroll
        for (int j = 0; j < 8; ++j) {
            float d = Xt[lr * HID + sc + j] - mu;
            pv += d * d;
        }
        red[tid] = pv;
        __syncthreads();
        if (tid < 16) {
            float v = 0.f;
            for (int k = 0; k < 16; ++k) v += red[tid * 16 + k];
            rs_s[tid] = rsqrtf(v * (1.0f / HID) + LN_EPS);
        }
        __syncthreads();

#pragma unroll
        for (int j = 0; j < 8; ++j) {
            int e = tid * 8 + j;
            int r = e >> 7, c = e & 127;
            float y = lng[c] * (Xt[e] - mu_s[r]) * rs_s[r] + lnb[c];
            h_io[(long)(row0 + r) * HID + c] = fmaxf(y, 0.0f);
        }
        __syncthreads();   // protect mu_s/rs_s/red before next tile reuses them
    }
}

// predictor: out = relu(h_net @ W1 + b1) @ W2 + b2
__global__ __launch_bounds__(256)
void predictor_kernel(const float* __restrict__ h_net,
                      const float* __restrict__ W1, const float* __restrict__ b1,
                      const float* __restrict__ W2, const float* __restrict__ b2,
                      float* __restrict__ out) {
    __shared__ float At[16 * HID];
    __shared__ float Ht[16 * HID];
    const int tid  = threadIdx.x;
    const int row0 = blockIdx.x * 16;

#pragma unroll
    for (int j = 0; j < 8; ++j) {
        int e = tid + j * 256;
        At[e] = h_net[(long)row0 * HID + e];
    }
    __syncthreads();

    const int wave = tid >> 5, lane = tid & 31;
    const int m = lane & 15, half = lane >> 4, koff = half * 2;
    const int col = wave * 16 + m;

    v8f acc = {};
#pragma unroll
    for (int kb = 0; kb < HID; kb += 4) {
        v2f a, b;
        a.x = At[m * HID + kb + koff];
        a.y = At[m * HID + kb + koff + 1];
        b.x = W1[(long)(kb + koff) * HID + col];
        b.y = W1[(long)(kb + koff + 1) * HID + col];
        acc = __builtin_amdgcn_wmma_f32_16x16x4_f32(false, a, false, b,
                                                    (short)0, acc, false, false);
    }
    const float bb = b1[col];
#pragma unroll
    for (int i = 0; i < 8; ++i) {
        int r = i + half * 8;
        Ht[r * HID + col] = fmaxf(acc[i] + bb, 0.0f);
    }
    __syncthreads();

    if (tid < 16) {
        float s = 0.f;
        for (int c = 0; c < HID; ++c) s += Ht[tid * HID + c] * W2[c];
        out[row0 + tid] = s + b2[0];
    }
}

// ---------------------------------------------------------------------------
extern "C" void kernel_launch(void* const* d_in, const int* in_sizes, int n_in,
                              void* d_out, int out_size, void* d_ws, size_t ws_size,
                              hipStream_t stream) {
    (void)in_sizes; (void)n_in; (void)out_size; (void)ws_size;

    const float* pin_feats = (const float*)d_in[0];
    const int*   p2n_src   = (const int*)d_in[1];
    const int*   p2n_dst   = (const int*)d_in[2];
    const int*   n2n_src   = (const int*)d_in[3];
    const int*   n2n_dst   = (const int*)d_in[4];
    const int*   n2p_src   = (const int*)d_in[5];
    const int*   n2p_dst   = (const int*)d_in[6];
    const float* ow        = (const float*)d_in[7];
    const float* Wproj     = (const float*)d_in[8];
    const float* bproj     = (const float*)d_in[9];
    const float* Wp2n      = (const float*)d_in[10];
    const float* bp2n      = (const float*)d_in[11];
    const float* Wn2n      = (const float*)d_in[12];
    const float* bn2n      = (const float*)d_in[13];
    const float* Wn2p      = (const float*)d_in[14];
    const float* bn2p      = (const float*)d_in[15];
    const float* ln1_g     = (const float*)d_in[16];
    const float* ln1_b     = (const float*)d_in[17];
    const float* ln2_g     = (const float*)d_in[18];
    const float* ln2_b     = (const float*)d_in[19];
    const float* lnp_g     = (const float*)d_in[20];
    const float* lnp_b     = (const float*)d_in[21];
    const float* W1        = (const float*)d_in[22];
    const float* b1        = (const float*)d_in[23];
    const float* W2        = (const float*)d_in[24];
    const float* b2        = (const float*)d_in[25];

    // ---------------- workspace layout -------------------------------------
    float* fp = (float*)d_ws;
    float* h_pin    = fp; fp += (size_t)N_PIN * HID;
    float* h_net    = fp; fp += (size_t)N_NET * HID;
    float* agg_net  = fp; fp += (size_t)N_NET * HID;
    float* agg_pin  = fp; fp += (size_t)N_PIN * HID;
    float* csrw_n2n = fp; fp += (size_t)E_N2N;
    int* ip = (int*)fp;
    int* cnt_p2n  = ip; ip += N_NET;
    int* offs_p2n = ip; ip += N_NET;
    int* cur_p2n  = ip; ip += N_NET;
    int* csr_p2n  = ip; ip += E_P2N;
    int* cnt_n2n  = ip; ip += N_NET;
    int* offs_n2n = ip; ip += N_NET;
    int* cur_n2n  = ip; ip += N_NET;
    int* csr_n2n  = ip; ip += E_N2N;
    int* cnt_n2p  = ip; ip += N_PIN;
    int* offs_n2p = ip; ip += N_PIN;
    int* cur_n2p  = ip; ip += N_PIN;
    int* csr_n2p  = ip; ip += E_P2N;
    int* bsum     = ip; ip += 1024;

    // ---------------- build CSR by destination (per call) ------------------
    hipMemsetAsync(cnt_p2n, 0, (size_t)N_NET * sizeof(int), stream);
    hipMemsetAsync(cnt_n2n, 0, (size_t)N_NET * sizeof(int), stream);
    hipMemsetAsync(cnt_n2p, 0, (size_t)N_PIN * sizeof(int), stream);
    count_kernel<<<(E_P2N + 255) / 256, 256, 0, stream>>>(p2n_dst, E_P2N, cnt_p2n);
    count_kernel<<<(E_N2N + 255) / 256, 256, 0, stream>>>(n2n_dst, E_N2N, cnt_n2n);
    count_kernel<<<(E_P2N + 255) / 256, 256, 0, stream>>>(n2p_dst, E_P2N, cnt_n2p);

    auto scan = [&](const int* cnt, int n, int* offs) {
        const int nb = (n + 255) / 256;
        scan_block_kernel<<<nb, 256, 0, stream>>>(cnt, n, offs, bsum);
        scan_bsum_kernel<<<1, 1, 0, stream>>>(bsum, nb);
        scan_add_kernel<<<nb, 256, 0, stream>>>(offs, bsum, n);
    };
    scan(cnt_p2n, N_NET, offs_p2n);
    scan(cnt_n2n, N_NET, offs_n2n);
    scan(cnt_n2p, N_PIN, offs_n2p);

    hipMemcpyAsync(cur_p2n, offs_p2n, (size_t)N_NET * sizeof(int),
                   hipMemcpyDeviceToDevice, stream);
    hipMemcpyAsync(cur_n2n, offs_n2n, (size_t)N_NET * sizeof(int),
                   hipMemcpyDeviceToDevice, stream);
    hipMemcpyAsync(cur_n2p, offs_n2p, (size_t)N_PIN * sizeof(int),
                   hipMemcpyDeviceToDevice, stream);

    csr_fill_kernel<<<(E_P2N + 255) / 256, 256, 0, stream>>>(
        p2n_src, p2n_dst, nullptr, E_P2N, cur_p2n, csr_p2n, nullptr);
    csr_fill_kernel<<<(E_N2N + 255) / 256, 256, 0, stream>>>(
        n2n_src, n2n_dst, ow, E_N2N, cur_n2n, csr_n2n, csrw_n2n);
    csr_fill_kernel<<<(E_P2N + 255) / 256, 256, 0, stream>>>(
        n2p_src, n2p_dst, nullptr, E_P2N, cur_n2p, csr_n2p, nullptr);

    // ---------------- forward pass -----------------------------------------
    hipMemsetAsync(h_net, 0, (size_t)N_NET * HID * sizeof(float), stream);
    proj_kernel<<<N_PIN / 16, 256, 0, stream>>>(pin_feats, Wproj, bproj, h_pin);

    const int gnet = (N_NET / 16) / RTILES;   // 625
    const int gpin = (N_PIN / 16) / RTILES;   // 2500

    for (int l = 0; l < NL; ++l) {
        const size_t wo = (size_t)l * HID * HID;
        const size_t vo = (size_t)l * HID;
        // stage 1: pin -> net
        gather_kernel<<<N_NET / 8, 256, 0, stream>>>(h_pin, offs_p2n, cnt_p2n,
                                                     csr_p2n, nullptr, agg_net);
        gconv_finish_kernel<<<gnet, 256, 0, stream>>>(
            agg_net, Wp2n + wo, bp2n + vo, h_net, ln1_g + vo, ln1_b + vo);
        // stage 2: net -> net (edge weighted)
        gather_kernel<<<N_NET / 8, 256, 0, stream>>>(h_net, offs_n2n, cnt_n2n,
                                                     csr_n2n, csrw_n2n, agg_net);
        gconv_finish_kernel<<<gnet, 256, 0, stream>>>(
            agg_net, Wn2n + wo, bn2n + vo, h_net, ln2_g + vo, ln2_b + vo);
        // stage 3: net -> pin
        gather_kernel<<<N_PIN / 8, 256, 0, stream>>>(h_net, offs_n2p, cnt_n2p,
                                                     csr_n2p, nullptr, agg_pin);
        gconv_finish_kernel<<<gpin, 256, 0, stream>>>(
            agg_pin, Wn2p + wo, bn2p + vo, h_pin, lnp_g + vo, lnp_b + vo);
    }

    // ---------------- prediction head + h_net output -----------------------
    float* out = (float*)d_out;
    predictor_kernel<<<N_NET / 16, 256, 0, stream>>>(h_net, W1, b1, W2, b2, out);
    hipMemcpyAsync(out + N_NET, h_net, (size_t)N_NET * HID * sizeof(float),
                   hipMemcpyDeviceToDevice, stream);
}